// MultimodalAttention_39178691674269
// MI455X (gfx1250) — compile-verified
//
#include <hip/hip_runtime.h>
#include <hip/hip_bf16.h>

typedef __attribute__((ext_vector_type(2))) float v2f;
typedef __attribute__((ext_vector_type(8))) float v8f;

#define MM_D      128
#define MM_MODAL  4
#define MM_ALPHA  0.05f
#define MM_EPS    1e-5f

// ---------------------------------------------------------------------------
// Kernel 1: zero the aggregation buffer (d_out doubles as agg scratch)
// ---------------------------------------------------------------------------
__global__ __launch_bounds__(256) void mm_zero_kernel(float* __restrict__ agg,
                                                      long long n4) {
    long long i = (long long)blockIdx.x * 256 + threadIdx.x;
    if (i < n4) {
        ((float4*)agg)[i] = make_float4(0.f, 0.f, 0.f, 0.f);
    }
}

// ---------------------------------------------------------------------------
// Kernel 2: edge gather->weight->scatter-add.  128 consecutive threads handle
// one edge row (512 floats, 4 per thread): b128 gathers + coalesced f32
// atomics (global_atomic_add_f32).
// ---------------------------------------------------------------------------
__global__ __launch_bounds__(256) void mm_scatter_kernel(
    const float* __restrict__ x,        // [N, 512]
    const int*   __restrict__ esrc,
    const int*   __restrict__ edst,
    const float* __restrict__ ew,
    float*       __restrict__ agg,      // [N, 512]
    long long total)                    // E * 128  (one thread = 4 floats)
{
    long long tid = (long long)blockIdx.x * 256 + threadIdx.x;
    if (tid >= total) return;
    int e  = (int)(tid >> 7);           // edge id
    int c4 = ((int)tid & 127) << 2;     // float offset within row (0,4,...,508)
    float w = ew[e];
    int   s = esrc[e];
    int   d = edst[e];
    float4 xv = *(const float4*)&x[(size_t)s * 512 + c4];
    float* dp = &agg[(size_t)d * 512 + c4];
    atomicAdd(dp + 0, w * xv.x);
    atomicAdd(dp + 1, w * xv.y);
    atomicAdd(dp + 2, w * xv.z);
    atomicAdd(dp + 3, w * xv.w);
}

// ---------------------------------------------------------------------------
// Kernel 3: fused  out = LayerNorm(x + alpha * (agg @ W)) using
// V_WMMA_F32_16X16X4_F32.  Each wave owns a 16-row x 128-col strip.
// W is staged in LDS as interleaved K-pairs so every B operand is one
// aligned ds_load_b64 directly into an even VGPR pair.
// In-place on `agg` buffer (each wave reads only the rows it writes).
// ---------------------------------------------------------------------------
__global__ __launch_bounds__(256) void mm_gemm_ln_kernel(
    const float* __restrict__ x,        // [R, 128] residual input (multimodal)
    const float* __restrict__ W,        // [128, 128]
    const float* __restrict__ gamma,    // [128]
    const float* __restrict__ beta,     // [128]
    float*       __restrict__ out,      // in: agg [R,128]; out: result [R,128]
    int R)
{
    // Wl[kp*128 + col] = { W[2kp][col], W[2kp+1][col] }   (64 KB)
    __shared__ v2f Wl[(MM_D / 2) * MM_D];

    const int tid = threadIdx.x;
    for (int i = tid; i < (MM_D / 2) * MM_D; i += 256) {
        const int kp  = i >> 7;         // pair-row 0..63
        const int col = i & 127;
        v2f p;
        p.x = W[(2 * kp + 0) * MM_D + col];
        p.y = W[(2 * kp + 1) * MM_D + col];
        Wl[i] = p;
    }
    __syncthreads();

    const int wave = tid >> 5;                    // 0..7
    const int lane = tid & 31;
    const int r0   = (blockIdx.x * 8 + wave) * 16;
    if (r0 >= R) return;                          // wave-uniform exit (R % 16 == 0)

    const int lrow = lane & 15;                   // 0..15
    const int hgrp = lane >> 4;                   // half-wave id: 0 or 1
    const int kb   = hgrp * 2;                    // K sub-offset for A/B operands

    v8f acc[8];
#pragma unroll
    for (int t = 0; t < 8; ++t)
#pragma unroll
        for (int v = 0; v < 8; ++v) acc[t][v] = 0.f;

    const float* arow = out + (size_t)(r0 + lrow) * MM_D;

    // K loop: 32 steps of K=4, 8 WMMAs per step (one per 16-col tile)
    for (int k0 = 0; k0 < MM_D; k0 += 4) {
        // A(16x4 f32): lane holds M=lane%16, K = k0 + 2*hgrp + {0,1}
        v2f a = *(const v2f*)&arow[k0 + kb];      // one global_load_b64
        const int kp = (k0 >> 1) + hgrp;          // LDS pair-row
#pragma unroll
        for (int t = 0; t < 8; ++t) {
            // B(4x16): lane holds N=lane%16, K = k0 + 2*hgrp + {0,1}
            v2f b = Wl[kp * MM_D + t * 16 + lrow];  // one ds_load_b64
            // 8 args: (neg_a, A, neg_b, B, c_mod, C, reuse_a, reuse_b)
            acc[t] = __builtin_amdgcn_wmma_f32_16x16x4_f32(
                false, a, false, b, (short)0, acc[t], false, false);
        }
    }

    // ---- residual + LayerNorm ------------------------------------------------
    // C/D layout: acc[t][v] = out1[row = r0 + v + 8*hgrp][col = 16t + lrow]
    float s1[8], s2[8];
#pragma unroll
    for (int v = 0; v < 8; ++v) { s1[v] = 0.f; s2[v] = 0.f; }

#pragma unroll
    for (int t = 0; t < 8; ++t) {
        const int col = t * 16 + lrow;
#pragma unroll
        for (int v = 0; v < 8; ++v) {
            const size_t row = (size_t)(r0 + 8 * hgrp + v);
            float yy = x[row * MM_D + col] + MM_ALPHA * acc[t][v];
            acc[t][v] = yy;                       // reuse accumulators as y
            s1[v] += yy;
            s2[v] += yy * yy;
        }
    }

    // butterfly reduce across the 16 lanes holding each row
    // (xor masks 1,2,4,8 keep lanes inside their 16-lane half of the wave32)
#pragma unroll
    for (int v = 0; v < 8; ++v) {
#pragma unroll
        for (int m = 1; m <= 8; m <<= 1) {
            s1[v] += __shfl_xor(s1[v], m, 32);
            s2[v] += __shfl_xor(s2[v], m, 32);
        }
    }

    float mean[8], rstd[8];
    const float invD = 1.0f / (float)MM_D;
#pragma unroll
    for (int v = 0; v < 8; ++v) {
        float mu  = s1[v] * invD;
        float var = s2[v] * invD - mu * mu;
        mean[v] = mu;
        rstd[v] = __frsqrt_rn(var + MM_EPS);
    }

    float g[8], bb[8];
#pragma unroll
    for (int t = 0; t < 8; ++t) {
        const int col = t * 16 + lrow;
        g[t]  = gamma[col];
        bb[t] = beta[col];
    }

#pragma unroll
    for (int t = 0; t < 8; ++t) {
        const int col = t * 16 + lrow;
#pragma unroll
        for (int v = 0; v < 8; ++v) {
            const size_t row = (size_t)(r0 + 8 * hgrp + v);
            out[row * MM_D + col] =
                (acc[t][v] - mean[v]) * rstd[v] * g[t] + bb[t];
        }
    }
}

// ---------------------------------------------------------------------------
extern "C" void kernel_launch(void* const* d_in, const int* in_sizes, int n_in,
                              void* d_out, int out_size, void* d_ws, size_t ws_size,
                              hipStream_t stream) {
    const float* multimodal = (const float*)d_in[0];   // [N,4,128]
    const int*   edge_src   = (const int*)  d_in[1];   // [E]
    const int*   edge_dst   = (const int*)  d_in[2];   // [E]
    const float* edge_w     = (const float*)d_in[3];   // [E]
    const float* W          = (const float*)d_in[4];   // [128,128]
    const float* gamma      = (const float*)d_in[5];   // [128]
    const float* beta       = (const float*)d_in[6];   // [128]
    float*       out        = (float*)d_out;

    const long long E = in_sizes[1];
    const long long totalElems = (long long)in_sizes[0];     // N*4*128
    const int R = (int)(totalElems / MM_D);                  // N*4 rows

    // 1) zero agg (aliases d_out; kernel 3 fully overwrites it)
    {
        long long n4 = totalElems / 4;
        int grid = (int)((n4 + 255) / 256);
        mm_zero_kernel<<<grid, 256, 0, stream>>>(out, n4);
    }
    // 2) edge scatter-add (4 floats per thread)
    {
        long long total = E * (long long)(MM_MODAL * MM_D / 4);  // E*128
        long long grid  = (total + 255) / 256;
        mm_scatter_kernel<<<(unsigned)grid, 256, 0, stream>>>(
            multimodal, edge_src, edge_dst, edge_w, out, total);
    }
    // 3) fused WMMA GEMM + residual + LayerNorm (in place on d_out)
    {
        int grid = (R + 127) / 128;                          // 128 rows / block
        mm_gemm_ln_kernel<<<grid, 256, 0, stream>>>(
            multimodal, W, gamma, beta, out, R);
    }
}